// Model_87076166959211
// MI455X (gfx1250) — compile-verified
//
#include <hip/hip_runtime.h>
#include <math.h>
#include <stdint.h>

#define TI 2048   // input_size
#define TH 2048   // hidden_size
#define TT 512    // window_size
#define TL 4      // layers

typedef __attribute__((ext_vector_type(2))) float v2f;
typedef __attribute__((ext_vector_type(8))) float v8f;
typedef __attribute__((ext_vector_type(4))) unsigned int u32x4;
typedef __attribute__((ext_vector_type(4))) int i32x4;
typedef __attribute__((ext_vector_type(8))) int i32x8;

#if defined(__has_builtin)
#if __has_builtin(__builtin_amdgcn_tensor_load_to_lds) && \
    __has_builtin(__builtin_amdgcn_s_wait_tensorcnt)
#define USE_TDM 1
#endif
#endif
#ifndef USE_TDM
#define USE_TDM 0
#endif

// ---------------------------------------------------------------------------
// Kernel 1: pre[M=T][N=H] = A[M][K] @ W[N][K]^T + bias[N]   (fp32 WMMA 16x16x4)
// A accessed with generic strides so layer 0 can read x (I,T) transposed.
// 256 threads = 8 waves as 4(M) x 2(N). Tile: BM=64 x BN=128, K-chunk BK=32.
// Each wave computes a 16x64 strip = 4 fragments sharing one A fragment.
// B tile (BN rows x BK floats, row pitch 8KB in W) is DMA'd by the Tensor
// Data Mover into LDS (double buffered, TENSORcnt-pipelined); LDS row pitch
// is padded to 34 floats by the TDM pad feature for conflict-free reads.
// ---------------------------------------------------------------------------
#define BM 64
#define BN 128
#define BK 32
#define BPITCH 34   // BK + 2 DWORD pad inserted by TDM every 32 DWORDs

#if USE_TDM
// Issue one TDM 2D tile load: 128 rows x 32 floats from gsrc (row stride 2048
// floats) into LDS at lds_off, padding +2 DWORDs after every 32 DWORDs.
__device__ __forceinline__ void tdm_load_b_tile(const float* gsrc,
                                                unsigned lds_off) {
  const unsigned long long ga = (unsigned long long)(uintptr_t)gsrc;
  u32x4 g0;
  g0[0] = 1u;                                   // count=1, user descriptor
  g0[1] = lds_off;                              // LDS byte address
  g0[2] = (unsigned)(ga & 0xFFFFFFFFu);         // global_addr[31:0]
  g0[3] = (unsigned)((ga >> 32) & 0x1FFFFFFu)   // global_addr[56:32]
          | (2u << 30);                         // type = 2 ("image")
  i32x8 g1;
  g1[0] = (2 << 16)      // data_size = 4 bytes
        | (1 << 20)      // pad_enable
        | (4 << 22)      // pad_interval code 4 -> every 32 DWORDs
        | (1 << 25);     // pad_amount  code 1 -> 2 DWORDs
  g1[1] = (BK << 16);            // tensor_dim0[15:0] in bits 79:48 (low half)
  g1[2] = (BN << 16);            // tensor_dim0 hi=0 | tensor_dim1[15:0]
  g1[3] = (BK << 16);            // tensor_dim1 hi=0 | tile_dim0
  g1[4] = BN;                    // tile_dim1 | tile_dim2=0
  g1[5] = TH;                    // tensor_dim0_stride[31:0] = 2048 elements
  g1[6] = 0;                     // stride0 hi | tensor_dim1_stride lo
  g1[7] = 0;
  const i32x4 z4 = {0, 0, 0, 0};           // groups 2/3 unused (2D tensor)
  const i32x8 z8 = {0, 0, 0, 0, 0, 0, 0, 0};
  __builtin_amdgcn_tensor_load_to_lds(g0, g1, z4, z4, z8, 0);
}
#endif

__global__ __launch_bounds__(256)
void rnn_pre_gemm(const float* __restrict__ A, int a_rs, int a_cs,
                  const float* __restrict__ W,
                  const float* __restrict__ bias,
                  float* __restrict__ out)
{
  __shared__ float As[BM][BK + 1];
  __shared__ float Bs[2][BN][BPITCH];   // [n][k] with TDM pad pitch

  const int tid  = threadIdx.x;
  const int lane = tid & 31;
  const int wid  = tid >> 5;
  const int wm   = wid & 3;        // 0..3 : M sub-tile (16 rows each)
  const int wn   = wid >> 2;       // 0..1 : N sub-tile (64 cols each)
  const int half = lane >> 4;      // 0: lanes 0-15, 1: lanes 16-31
  const int l16  = lane & 15;

  const int tileM = (blockIdx.x & 7) * BM;    // 512/64  = 8 M tiles
  const int tileN = (blockIdx.x >> 3) * BN;   // 2048/128 = 16 N tiles

  v8f acc[4];
#pragma unroll
  for (int f = 0; f < 4; ++f) {
    const float bv = bias[tileN + wn * 64 + f * 16 + l16];
    acc[f] = (v8f){bv, bv, bv, bv, bv, bv, bv, bv};
  }

#if USE_TDM
  // Prologue: DMA chunk 0 into buffer 0 (one wave issues; EXEC-independent).
  if (wid == 0)
    tdm_load_b_tile(W + (size_t)tileN * TH,
                    (unsigned)(uintptr_t)&Bs[0][0][0]);
#endif

  for (int k0 = 0; k0 < TH; k0 += BK) {
    const int buf = (k0 / BK) & 1;

    // ---- stage A tile (BM x BK = 2048 elems, 8 per thread) ----
#pragma unroll
    for (int i = 0; i < 8; ++i) {
      const int e = tid + i * 256;
      const int r = e >> 5;
      const int c = e & 31;
      As[r][c] = A[(size_t)(tileM + r) * a_rs + (size_t)(k0 + c) * a_cs];
    }

#if USE_TDM
    if (wid == 0) {
      if (k0 + BK < TH) {
        // issue next chunk into the other buffer, then retire current chunk
        tdm_load_b_tile(W + (size_t)tileN * TH + (k0 + BK),
                        (unsigned)(uintptr_t)&Bs[buf ^ 1][0][0]);
        __builtin_amdgcn_s_wait_tensorcnt(1);
      } else {
        __builtin_amdgcn_s_wait_tensorcnt(0);
      }
    }
#else
    // Fallback: stage B with vector loads. Bs[buf][n][k] = W[tileN+n][k0+k].
    {
      const int nrow = tid >> 3;        // 0..31
      const int kb   = (tid & 7) * 4;   // 0,4,...,28
#pragma unroll
      for (int p = 0; p < 4; ++p) {
        const int n = nrow + p * 32;
        const float4 w4 =
            *(const float4*)(W + (size_t)(tileN + n) * TH + (k0 + kb));
        Bs[buf][n][kb + 0] = w4.x;
        Bs[buf][n][kb + 1] = w4.y;
        Bs[buf][n][kb + 2] = w4.z;
        Bs[buf][n][kb + 3] = w4.w;
      }
    }
#endif
    __syncthreads();   // As + Bs[buf] visible to all waves

    // ---- 8 WMMA k-steps of 4 ----
#pragma unroll
    for (int kk = 0; kk < BK; kk += 4) {
      // A fragment 16x4: lanes 0-15 hold K={kk,kk+1}, lanes 16-31 K={kk+2,kk+3}
      v2f a;
      a.x = As[wm * 16 + l16][kk + 2 * half];
      a.y = As[wm * 16 + l16][kk + 2 * half + 1];
#pragma unroll
      for (int f = 0; f < 4; ++f) {
        const int nn = wn * 64 + f * 16 + l16;
        v2f b;
        b.x = Bs[buf][nn][kk + 2 * half];
        b.y = Bs[buf][nn][kk + 2 * half + 1];
        acc[f] = __builtin_amdgcn_wmma_f32_16x16x4_f32(
            /*neg_a=*/false, a, /*neg_b=*/false, b,
            /*c_mod=*/(short)0, acc[f], /*reuse_a=*/false, /*reuse_b=*/false);
      }
    }
    __syncthreads();   // compute done: As and Bs[buf] may be overwritten
  }

  // ---- store: VGPR r holds M = r + 8*half, N = lane&15 ----
#pragma unroll
  for (int f = 0; f < 4; ++f) {
    const int col = tileN + wn * 64 + f * 16 + l16;
#pragma unroll
    for (int r = 0; r < 8; ++r) {
      const int row = tileM + wm * 16 + half * 8 + r;
      out[(size_t)row * TH + col] = acc[f][r];
    }
  }
}

// ---------------------------------------------------------------------------
// Kernel 2: one recurrence step. One wave32 per output row.
// h[t][row] = tanh(pre[t][row] + dot(W_hh[row,:], h[t-1,:]) + b_hh[row])
// W_hh (16 MB) and h vectors stay L2-resident across all 512 steps.
// ---------------------------------------------------------------------------
__global__ __launch_bounds__(256)
void rnn_step(const float* __restrict__ pre,
              const float* __restrict__ whh,
              const float* __restrict__ bhh,
              float* __restrict__ hseq,
              int t)
{
  const int lane = threadIdx.x & 31;
  const int row  = (int)((blockIdx.x * blockDim.x + threadIdx.x) >> 5);
  if (row >= TH) return;

  float sum = 0.f;
  if (t > 0) {
    const float4* wrow  = (const float4*)(whh + (size_t)row * TH);
    const float4* hprev = (const float4*)(hseq + (size_t)(t - 1) * TH);
#pragma unroll 4
    for (int i = lane; i < TH / 4; i += 32) {
      const float4 w = wrow[i];
      const float4 h = hprev[i];
      sum += w.x * h.x + w.y * h.y + w.z * h.z + w.w * h.w;
    }
  }
#pragma unroll
  for (int off = 16; off > 0; off >>= 1)
    sum += __shfl_xor(sum, off, 32);

  if (lane == 0)
    hseq[(size_t)t * TH + row] =
        tanhf(pre[(size_t)t * TH + row] + sum + bhh[row]);
}

// ---------------------------------------------------------------------------
// Kernel 3: final FC on last timestep. out[i] = dot(w_fc[i,:], h_last) + b_fc[i]
// ---------------------------------------------------------------------------
__global__ __launch_bounds__(256)
void fc_kernel(const float* __restrict__ hlast,
               const float* __restrict__ wfc,
               const float* __restrict__ bfc,
               float* __restrict__ out)
{
  const int lane = threadIdx.x & 31;
  const int row  = (int)((blockIdx.x * blockDim.x + threadIdx.x) >> 5);
  if (row >= TI) return;

  const float4* w = (const float4*)(wfc + (size_t)row * TH);
  const float4* h = (const float4*)hlast;
  float sum = 0.f;
#pragma unroll 4
  for (int i = lane; i < TH / 4; i += 32) {
    const float4 a = w[i];
    const float4 b = h[i];
    sum += a.x * b.x + a.y * b.y + a.z * b.z + a.w * b.w;
  }
#pragma unroll
  for (int off = 16; off > 0; off >>= 1)
    sum += __shfl_xor(sum, off, 32);

  if (lane == 0) out[row] = sum + bfc[row];
}

// ---------------------------------------------------------------------------
extern "C" void kernel_launch(void* const* d_in, const int* in_sizes, int n_in,
                              void* d_out, int out_size, void* d_ws, size_t ws_size,
                              hipStream_t stream)
{
  const float* x    = (const float*)d_in[0];  // (I, T)
  const float* w_ih = (const float*)d_in[1];  // (L, H, H)
  const float* w_hh = (const float*)d_in[2];  // (L, H, H)
  const float* b_ih = (const float*)d_in[3];  // (L, H)
  const float* b_hh = (const float*)d_in[4];  // (L, H)
  const float* w_fc = (const float*)d_in[5];  // (I, H)
  const float* b_fc = (const float*)d_in[6];  // (I,)
  float* out = (float*)d_out;

  // workspace layout: pre (T*H) | hA (T*H) | hB (T*H)  ~= 12.6 MB
  float* pre = (float*)d_ws;
  float* hA  = pre + (size_t)TT * TH;
  float* hB  = hA + (size_t)TT * TH;

  const float* srcA = x;   // layer 0 reads x transposed: A[t][k] = x[k*T + t]
  int a_rs = 1, a_cs = TT;
  const float* hprev = nullptr;

  for (int l = 0; l < TL; ++l) {
    float* hout = (l & 1) ? hB : hA;

    rnn_pre_gemm<<<dim3(8 * 16), dim3(256), 0, stream>>>(
        srcA, a_rs, a_cs,
        w_ih + (size_t)l * TH * TH, b_ih + (size_t)l * TH, pre);

    for (int t = 0; t < TT; ++t) {
      rnn_step<<<dim3((TH * 32) / 256), dim3(256), 0, stream>>>(
          pre, w_hh + (size_t)l * TH * TH, b_hh + (size_t)l * TH, hout, t);
    }

    srcA = hout;
    a_rs = TH;
    a_cs = 1;
    hprev = hout;
  }

  fc_kernel<<<dim3((TI * 32) / 256), dim3(256), 0, stream>>>(
      hprev + (size_t)(TT - 1) * TH, w_fc, b_fc, out);
}